// WeightedBCE2d_26989574488433
// MI455X (gfx1250) — compile-verified
//
#include <hip/hip_runtime.h>
#include <hip/hip_bf16.h>

typedef float v2f __attribute__((ext_vector_type(2)));
typedef float v8f __attribute__((ext_vector_type(8)));

// -100/ln2 : BCE log-clamp expressed in the log2 domain
#define LOG2_CLAMP (-144.26950408889634f)
// -ln2 : converts accumulated log2-domain sum back to natural-log BCE
#define NEG_LN2 (-0.6931471805599453f)

// ---------------------------------------------------------------------------
// Wave32 full-lane sum on the CDNA5 matrix pipe.
// A (16x4 f32, 2 VGPRs): lane L, VGPR0 -> (M = L%16, K = L<16 ? 0 : 2).
// v in a.x, 0 in a.y, B = all ones  =>  D[m][n] = v_m + v_{m+16}.
// D VGPR j: lanes 0-15 = s_j, lanes 16-31 = s_{j+8}; summing d0..d7 leaves
// half-sums in each 16-lane group; one SWAPX16 ds_swizzle (imm 0x401F:
// xor=0x10, and=0x1F) folds the halves. EXEC must be all ones at call site.
// ---------------------------------------------------------------------------
__device__ __forceinline__ float wave_reduce_sum_wmma(float v) {
    v2f a; a.x = v;    a.y = 0.0f;
    v2f b; b.x = 1.0f; b.y = 1.0f;
    v8f c = {};
    v8f d = __builtin_amdgcn_wmma_f32_16x16x4_f32(
        /*neg_a=*/false, a, /*neg_b=*/false, b,
        /*c_mod=*/(short)0, c, /*reuse_a=*/false, /*reuse_b=*/false);
    float r = ((d[0] + d[1]) + (d[2] + d[3])) + ((d[4] + d[5]) + (d[6] + d[7]));
    r += __int_as_float(__builtin_amdgcn_ds_swizzle(__float_as_int(r), 0x401F));
    return r;
}

// Block reduction (blockDim.x == 256 -> 8 wave32s), deterministic tree.
__device__ __forceinline__ float block_reduce_sum(float v) {
    __shared__ float s[8];
    const int lane = threadIdx.x & 31;
    const int wid  = threadIdx.x >> 5;
    float w = wave_reduce_sum_wmma(v);          // all 256 threads active here
    if (lane == 0) s[wid] = w;
    __syncthreads();
    float total = 0.0f;
    if (wid == 0) {                             // whole wave 0 enters (uniform)
        float x = (lane < 8) ? s[lane] : 0.0f;  // select, not divergence
        total = wave_reduce_sum_wmma(x);
    }
    return total;                               // valid in wave 0
}

// Per-element contribution in the log2 domain, WITHOUT the -ln2 factor:
//   returns w * (t*log2(p) + (1-t)*log2(1-p))   (clamped at -100/ln2)
// Final result = (-ln2 / N) * sum(contributions).
// Branch-free: both select operands are computed unconditionally and the
// boolean OR is non-short-circuit, so everything lowers to v_cndmask.
__device__ __forceinline__ float bce_elem(float p, float t, float n) {
    float g = fmaxf(__builtin_amdgcn_logf(p),        LOG2_CLAMP); // v_log_f32
    float h = fmaxf(__builtin_amdgcn_logf(1.0f - p), LOG2_CLAMP);
    int   base_c    = (int)(t > 0.0f) | (int)(n == 0.0f);
    float wbase     = base_c ? 1.0f : 0.0f;          // cndmask of constants
    float one_plus_n = 1.0f + n;                     // computed unconditionally
    float w = (n > 0.0f) ? one_plus_n : wbase;       // cndmask of live values
    float b = __builtin_fmaf(t, g - h, h);           // t*g + (1-t)*h
    return w * b;
}

// Stage 1: streaming weighted-BCE partial sums. float4 (global_load_b128) path.
__global__ void __launch_bounds__(256)
wbce_partial_kernel(const float* __restrict__ p, const float* __restrict__ t,
                    const float* __restrict__ n, float* __restrict__ partial,
                    int nvec, int ntotal) {
    const float4* __restrict__ p4 = reinterpret_cast<const float4*>(p);
    const float4* __restrict__ t4 = reinterpret_cast<const float4*>(t);
    const float4* __restrict__ n4 = reinterpret_cast<const float4*>(n);

    const int gtid   = blockIdx.x * blockDim.x + threadIdx.x;
    const int stride = gridDim.x * blockDim.x;

    float acc = 0.0f;
    for (int i = gtid; i < nvec; i += stride) {
        float4 pv = p4[i];
        float4 tv = t4[i];
        float4 nv = n4[i];
        acc += bce_elem(pv.x, tv.x, nv.x);
        acc += bce_elem(pv.y, tv.y, nv.y);
        acc += bce_elem(pv.z, tv.z, nv.z);
        acc += bce_elem(pv.w, tv.w, nv.w);
    }
    // scalar tail (ntotal not divisible by 4)
    for (int i = 4 * nvec + gtid; i < ntotal; i += stride)
        acc += bce_elem(p[i], t[i], n[i]);

    float total = block_reduce_sum(acc);
    if (threadIdx.x == 0) partial[blockIdx.x] = total;
}

// Stage 2: single-block finisher; deterministic, applies (-ln2 / N).
__global__ void __launch_bounds__(256)
wbce_final_kernel(const float* __restrict__ partial, int nPartial,
                  float* __restrict__ out, float scale) {
    float acc = 0.0f;
    for (int i = threadIdx.x; i < nPartial; i += blockDim.x)
        acc += partial[i];
    float total = block_reduce_sum(acc);
    if (threadIdx.x == 0) out[0] = total * scale;
}

extern "C" void kernel_launch(void* const* d_in, const int* in_sizes, int n_in,
                              void* d_out, int out_size, void* d_ws, size_t ws_size,
                              hipStream_t stream) {
    const float* p = (const float*)d_in[0];   // input (probabilities)
    const float* t = (const float*)d_in[1];   // target
    const float* n = (const float*)d_in[2];   // negative_pixels
    float* out = (float*)d_out;
    float* partial = (float*)d_ws;

    const int N    = in_sizes[0];
    const int nvec = N / 4;

    const int threads = 256;
    int blocks = (nvec + threads - 1) / threads;
    if (blocks > 1024) blocks = 1024;          // 8192 wave32s: saturates HBM
    if (blocks < 1)    blocks = 1;

    wbce_partial_kernel<<<blocks, threads, 0, stream>>>(p, t, n, partial, nvec, N);
    wbce_final_kernel<<<1, threads, 0, stream>>>(partial, blocks, out,
                                                 NEG_LN2 / (float)N);
}